// WarpingOutput_23502061044048
// MI455X (gfx1250) — compile-verified
//
#include <hip/hip_runtime.h>

// ---- problem constants (fixed by the reference) ------------------------
#define HW_B 8
#define HW_H 512
#define HW_W 512
#define HW_C 3
#define NPIX (HW_B * HW_H * HW_W)   // 2,097,152 pixels
#define TILE 256                    // pixels per tile == blockDim.x
#define NTILES (NPIX / TILE)        // 8192, exact
#define GRID 2048                   // 4 tiles per block -> async pipeline depth

// ---- CDNA5 async global->LDS helpers -----------------------------------
// LDS offset of a __shared__ object: flat LDS addresses carry the byte
// offset in the low 32 bits (CDNA5 ISA aperture rules).
__device__ __forceinline__ unsigned lds_off(const void* p) {
  return (unsigned)(unsigned long long)p;
}

// GLOBAL_LOAD_ASYNC_TO_LDS_B64 (GV mode): vdst = LDS byte offset VGPR,
// vaddr = 64-bit global address VGPR pair. Tracked by ASYNCcnt.
// Flows are single-use streams -> TH_LOAD_NT so they don't evict the
// frames (which the gathers re-read heavily) from the 192 MB L2.
__device__ __forceinline__ void async_copy_b64_to_lds_nt(unsigned ldsoff, const void* gaddr) {
  asm volatile("global_load_async_to_lds_b64 %0, %1, off th:TH_LOAD_NT"
               :: "v"(ldsoff), "v"((unsigned long long)gaddr)
               : "memory");
}

template <int N>
__device__ __forceinline__ void wait_asynccnt() {
  asm volatile("s_wait_asynccnt %0" :: "i"(N) : "memory");
}

// ---- kernel -------------------------------------------------------------
__global__ __launch_bounds__(TILE) void warp4_kernel(
    const float* __restrict__ f0,  const float* __restrict__ f1,
    const float* __restrict__ f01, const float* __restrict__ f10,
    const float* __restrict__ ft0, const float* __restrict__ ft1,
    float* __restrict__ out)
{
  // double-buffered flow staging: [buf][which-flow][pixel-in-tile]
  __shared__ float2 sFlow[2][4][TILE];

  const int t = threadIdx.x;
  const float2* gflow0 = (const float2*)f01;
  const float2* gflow1 = (const float2*)f10;
  const float2* gflow2 = (const float2*)ft0;
  const float2* gflow3 = (const float2*)ft1;

  // prologue: async-stage flows for this block's first tile
  {
    const int p0 = blockIdx.x * TILE + t;
    async_copy_b64_to_lds_nt(lds_off(&sFlow[0][0][t]), (const void*)(gflow0 + p0));
    async_copy_b64_to_lds_nt(lds_off(&sFlow[0][1][t]), (const void*)(gflow1 + p0));
    async_copy_b64_to_lds_nt(lds_off(&sFlow[0][2][t]), (const void*)(gflow2 + p0));
    async_copy_b64_to_lds_nt(lds_off(&sFlow[0][3][t]), (const void*)(gflow3 + p0));
  }

  int j = 0;
  for (int cur = blockIdx.x; cur < NTILES; cur += GRID, ++j) {
    const int bufi = j & 1;
    const int nxt  = cur + GRID;

    if (nxt < NTILES) {
      // prefetch next tile's flows into the other LDS buffer, then wait
      // only for the CURRENT tile's 4 async loads (in-order completion).
      const int pn = nxt * TILE + t;
      const int nb = (j + 1) & 1;
      async_copy_b64_to_lds_nt(lds_off(&sFlow[nb][0][t]), (const void*)(gflow0 + pn));
      async_copy_b64_to_lds_nt(lds_off(&sFlow[nb][1][t]), (const void*)(gflow1 + pn));
      async_copy_b64_to_lds_nt(lds_off(&sFlow[nb][2][t]), (const void*)(gflow2 + pn));
      async_copy_b64_to_lds_nt(lds_off(&sFlow[nb][3][t]), (const void*)(gflow3 + pn));
      wait_asynccnt<4>();
    } else {
      wait_asynccnt<0>();
    }

    // ---- compute this tile (each thread = one pixel, all 4 warps) ----
    const int p   = cur * TILE + t;
    const int b   = p >> 18;               // / (512*512)
    const int rem = p & ((1 << 18) - 1);
    const int yi  = rem >> 9;              // / 512
    const int xi  = rem & 511;
    const float yf = (float)yi;
    const float xf = (float)xi;

    // warp k: image source and output section. out = concat(w(f1,f01),
    // w(f0,f10), w(f0,ft0), w(f1,ft1)) flat in return order.
    #pragma unroll
    for (int k = 0; k < 4; ++k) {
      const float* img = (k == 1 || k == 2) ? f0 : f1;
      const float2 fl  = sFlow[bufi][k][t];   // .x = dy, .y = dx

      float qy  = yf - fl.x;
      float qx  = xf - fl.y;
      float y0f = fminf(fmaxf(floorf(qy), 0.0f), (float)(HW_H - 2));
      float x0f = fminf(fmaxf(floorf(qx), 0.0f), (float)(HW_W - 2));
      float ay  = fminf(fmaxf(qy - y0f, 0.0f), 1.0f);
      float ax  = fminf(fmaxf(qx - x0f, 0.0f), 1.0f);
      int y0 = (int)y0f;
      int x0 = (int)x0f;

      // tl|tr are 6 contiguous floats, as are bl|br: two row-gathers.
      // These hit the frames, which we WANT resident in L2 (default RT).
      const float* pT = img + (((b * HW_H + y0) * HW_W) + x0) * HW_C;
      const float* pB = pT + HW_W * HW_C;
      float tl0 = pT[0], tl1 = pT[1], tl2 = pT[2];
      float tr0 = pT[3], tr1 = pT[4], tr2 = pT[5];
      float bl0 = pB[0], bl1 = pB[1], bl2 = pB[2];
      float br0 = pB[3], br1 = pB[4], br2 = pB[5];

      float top0 = fmaf(tr0 - tl0, ax, tl0);
      float top1 = fmaf(tr1 - tl1, ax, tl1);
      float top2 = fmaf(tr2 - tl2, ax, tl2);
      float bot0 = fmaf(br0 - bl0, ax, bl0);
      float bot1 = fmaf(br1 - bl1, ax, bl1);
      float bot2 = fmaf(br2 - bl2, ax, bl2);

      // Outputs are write-once: non-temporal stores keep them from
      // evicting the frames out of L2.
      float* po = out + (size_t)k * (size_t)NPIX * HW_C + (size_t)p * HW_C;
      __builtin_nontemporal_store(fmaf(bot0 - top0, ay, top0), po + 0);
      __builtin_nontemporal_store(fmaf(bot1 - top1, ay, top1), po + 1);
      __builtin_nontemporal_store(fmaf(bot2 - top2, ay, top2), po + 2);
    }
  }
}

// ---- host-side launch ---------------------------------------------------
extern "C" void kernel_launch(void* const* d_in, const int* in_sizes, int n_in,
                              void* d_out, int out_size, void* d_ws, size_t ws_size,
                              hipStream_t stream) {
  (void)in_sizes; (void)n_in; (void)out_size; (void)d_ws; (void)ws_size;
  const float* f0  = (const float*)d_in[0];  // frame_0
  const float* f1  = (const float*)d_in[1];  // frame_1
  const float* f01 = (const float*)d_in[2];
  const float* f10 = (const float*)d_in[3];
  const float* ft0 = (const float*)d_in[4];
  const float* ft1 = (const float*)d_in[5];
  float* out = (float*)d_out;

  hipLaunchKernelGGL(warp4_kernel, dim3(GRID), dim3(TILE), 0, stream,
                     f0, f1, f01, f10, ft0, ft1, out);
}